// GatedGCN_5626407158021
// MI455X (gfx1250) — compile-verified
//
#include <hip/hip_runtime.h>
#include <hip/hip_bf16.h>

#define DIM 128

typedef float v2f __attribute__((ext_vector_type(2)));
typedef float v8f __attribute__((ext_vector_type(8)));

// b128 async-copy operand type: int32x4 pointers in global (AS1) / LDS (AS3)
typedef int v4i_vs __attribute__((vector_size(4 * sizeof(int))));
typedef __attribute__((address_space(1))) v4i_vs as1_v4i;
typedef __attribute__((address_space(3))) v4i_vs as3_v4i;

#if defined(__gfx1250__) && __has_builtin(__builtin_amdgcn_global_load_async_to_lds_b128)
#define USE_ASYNC_LDS 1
#else
#define USE_ASYNC_LDS 0
#endif

// D = A(16x4) * B(4x16) + C(16x16), fp32 WMMA (CDNA5)
__device__ __forceinline__ v8f wmma_f32(v2f a, v2f b, v8f c) {
    return __builtin_amdgcn_wmma_f32_16x16x4_f32(
        /*neg_a=*/false, a, /*neg_b=*/false, b,
        /*c_mod=*/(short)0, c, /*reuse_a=*/false, /*reuse_b=*/false);
}

__device__ __forceinline__ float sigmoidf_(float x) {
    return 1.0f / (1.0f + __expf(-x));
}

// Copy 16B global -> LDS. Async (ASYNCcnt-tracked) path if the toolchain
// exposes it, otherwise classic global_load_b128 + ds_store_b128.
__device__ __forceinline__ void stage16(const float* __restrict__ g, float* l) {
#if USE_ASYNC_LDS
    __builtin_amdgcn_global_load_async_to_lds_b128(
        (as1_v4i*)g, (as3_v4i*)l, /*offset=*/0, /*cpol=*/0);
#else
    *(float4*)l = *(const float4*)g;
#endif
}

__device__ __forceinline__ void stage_wait() {
#if USE_ASYNC_LDS
#if __has_builtin(__builtin_amdgcn_s_wait_asynccnt)
    __builtin_amdgcn_s_wait_asynccnt(0);
#else
    asm volatile("s_wait_asynccnt 0" ::: "memory");
#endif
#endif
    __syncthreads();
}

// ---------------------------------------------------------------------------
// m = h @ W^T + b   (h: [N,128], W: [128,128] row-major as W[out][in])
// grid.x = N/16 row tiles; 256 threads = 8 waves; wave w -> col tile w*16.
// The 16x128 h row tile is staged in LDS once and shared by all 8 waves.
// ---------------------------------------------------------------------------
__global__ __launch_bounds__(256) void msg_gemm_kernel(
    const float* __restrict__ h, const float* __restrict__ W,
    const float* __restrict__ b, float* __restrict__ m) {
    __shared__ float s_h[16 * DIM];

    const int t = threadIdx.x;
    {
        const float* gsrc = h + (size_t)blockIdx.x * 16 * DIM;
        stage16(gsrc + t * 4,           s_h + t * 4);
        stage16(gsrc + (t + 256) * 4,   s_h + (t + 256) * 4);
    }
    stage_wait();

    const int wave  = t >> 5;
    const int lane  = t & 31;
    const int r16   = lane & 15;          // A row / B col / D col within tile
    const int half  = lane >> 4;          // 0: K=0,1   1: K=2,3
    const int koff  = half * 2;
    const int col   = wave * 16 + r16;

    v8f acc;
    const float bias = b[col];
#pragma unroll
    for (int v = 0; v < 8; ++v) acc[v] = bias;

    const float* __restrict__ lrow = s_h + r16 * DIM;
    const float* __restrict__ wrow = W + (size_t)col * DIM;   // B[k][n] = W[n][k]

#pragma unroll 8
    for (int k = 0; k < DIM; k += 4) {
        v2f a  = *(const v2f*)&lrow[k + koff];   // ds_load_b64
        v2f bb;
        bb.x = wrow[k + koff];  bb.y = wrow[k + koff + 1];
        acc = wmma_f32(a, bb, acc);
    }

#pragma unroll
    for (int v = 0; v < 8; ++v) {
        const int r = blockIdx.x * 16 + v + half * 8;
        m[(size_t)r * DIM + col] = acc[v];
    }
}

// ---------------------------------------------------------------------------
// agg[dst[e]] += m[src[e]]  (one wave per edge, float4 per lane, L2 atomics)
// ---------------------------------------------------------------------------
__global__ __launch_bounds__(256) void scatter_kernel(
    const float* __restrict__ m, const int* __restrict__ src,
    const int* __restrict__ dst, float* __restrict__ agg, int n_edges) {
    const int e    = blockIdx.x * 8 + (threadIdx.x >> 5);
    const int lane = threadIdx.x & 31;
    if (e >= n_edges) return;
    const int s = src[e];
    const int d = dst[e];
    const float4 val = *((const float4*)(m + (size_t)s * DIM) + lane);
    float* ap = agg + (size_t)d * DIM + lane * 4;
    atomicAdd(ap + 0, val.x);
    atomicAdd(ap + 1, val.y);
    atomicAdd(ap + 2, val.z);
    atomicAdd(ap + 3, val.w);
}

// ---------------------------------------------------------------------------
// Fused GRU step:
//   gi = agg @ Wih^T + bih ; gh = h @ Whh^T + bhh   (Wih/Whh: [384,128])
//   r = sig(ir+hr); z = sig(iz+hz); n = tanh(inn + r*hn)
//   h' = (1-z)*n + z*h   [+ optional ReLU]
// grid.x = N/16; 256 threads = 8 waves; wave w -> output cols w*16..w*16+15.
// agg & h row tiles staged in LDS (shared by all 8 waves); 6 WMMA
// accumulators per wave (r,z,n gates for input & hidden paths).
// ---------------------------------------------------------------------------
__global__ __launch_bounds__(256) void gru_kernel(
    const float* __restrict__ agg, const float* __restrict__ h,
    const float* __restrict__ Wih, const float* __restrict__ Whh,
    const float* __restrict__ bih, const float* __restrict__ bhh,
    float* __restrict__ h_out, int do_relu) {
    __shared__ float s_a[16 * DIM];
    __shared__ float s_h[16 * DIM];

    const int t = threadIdx.x;
    {
        const float* ga = agg + (size_t)blockIdx.x * 16 * DIM;
        const float* gh = h   + (size_t)blockIdx.x * 16 * DIM;
        stage16(ga + t * 4,           s_a + t * 4);
        stage16(ga + (t + 256) * 4,   s_a + (t + 256) * 4);
        stage16(gh + t * 4,           s_h + t * 4);
        stage16(gh + (t + 256) * 4,   s_h + (t + 256) * 4);
    }
    stage_wait();

    const int wave = t >> 5;
    const int lane = t & 31;
    const int r16  = lane & 15;
    const int half = lane >> 4;
    const int koff = half * 2;
    const int col  = wave * 16 + r16;

    v8f ir = {}, iz = {}, inn = {}, hr = {}, hz = {}, hn = {};

    const float* __restrict__ arow = s_a + r16 * DIM;
    const float* __restrict__ hrow = s_h + r16 * DIM;
    const float* __restrict__ wr = Wih + (size_t)(col)         * DIM;
    const float* __restrict__ wz = Wih + (size_t)(DIM + col)   * DIM;
    const float* __restrict__ wn = Wih + (size_t)(2*DIM + col) * DIM;
    const float* __restrict__ ur = Whh + (size_t)(col)         * DIM;
    const float* __restrict__ uz = Whh + (size_t)(DIM + col)   * DIM;
    const float* __restrict__ un = Whh + (size_t)(2*DIM + col) * DIM;

#pragma unroll 4
    for (int k = 0; k < DIM; k += 4) {
        v2f aa = *(const v2f*)&arow[k + koff];   // ds_load_b64
        v2f ah = *(const v2f*)&hrow[k + koff];   // ds_load_b64
        v2f b0, b1, b2, b3, b4, b5;
        b0.x = wr[k + koff];    b0.y = wr[k + koff + 1];
        b1.x = wz[k + koff];    b1.y = wz[k + koff + 1];
        b2.x = wn[k + koff];    b2.y = wn[k + koff + 1];
        b3.x = ur[k + koff];    b3.y = ur[k + koff + 1];
        b4.x = uz[k + koff];    b4.y = uz[k + koff + 1];
        b5.x = un[k + koff];    b5.y = un[k + koff + 1];
        ir  = wmma_f32(aa, b0, ir);
        iz  = wmma_f32(aa, b1, iz);
        inn = wmma_f32(aa, b2, inn);
        hr  = wmma_f32(ah, b3, hr);
        hz  = wmma_f32(ah, b4, hz);
        hn  = wmma_f32(ah, b5, hn);
    }

    const float br  = bih[col],         bhr = bhh[col];
    const float bz  = bih[DIM + col],   bhz = bhh[DIM + col];
    const float bn  = bih[2*DIM + col], bhn = bhh[2*DIM + col];

#pragma unroll
    for (int v = 0; v < 8; ++v) {
        const int rloc = v + half * 8;
        const int rg   = blockIdx.x * 16 + rloc;
        const float hv = s_h[rloc * DIM + col];
        const float r  = sigmoidf_(ir[v]  + br + hr[v] + bhr);
        const float z  = sigmoidf_(iz[v]  + bz + hz[v] + bhz);
        const float nn = tanhf(inn[v] + bn + r * (hn[v] + bhn));
        float o = (1.0f - z) * nn + z * hv;
        if (do_relu) o = fmaxf(o, 0.0f);
        h_out[(size_t)rg * DIM + col] = o;
    }
}

// ---------------------------------------------------------------------------
__global__ void zero_kernel(float* __restrict__ p, int n) {
    const int i = blockIdx.x * blockDim.x + threadIdx.x;
    if (i < n) p[i] = 0.0f;
}

// out[col] += scale * sum over a 256-row chunk  (128 threads = 128 cols)
__global__ void mean_kernel(const float* __restrict__ h, float* __restrict__ out,
                            int n_rows, float scale) {
    const int col  = threadIdx.x;
    const int row0 = blockIdx.x * 256;
    const int rend = min(row0 + 256, n_rows);
    float s = 0.0f;
    for (int r = row0; r < rend; ++r) s += h[(size_t)r * DIM + col];
    atomicAdd(&out[col], s * scale);
}

// ---------------------------------------------------------------------------
extern "C" void kernel_launch(void* const* d_in, const int* in_sizes, int n_in,
                              void* d_out, int out_size, void* d_ws, size_t ws_size,
                              hipStream_t stream) {
    const float* in_feat = (const float*)d_in[0];
    const int*   src     = (const int*)d_in[1];
    const int*   dst     = (const int*)d_in[2];

    const float* Wm[2]  = {(const float*)d_in[3],  (const float*)d_in[9]};
    const float* bm[2]  = {(const float*)d_in[4],  (const float*)d_in[10]};
    const float* Wih[2] = {(const float*)d_in[5],  (const float*)d_in[11]};
    const float* Whh[2] = {(const float*)d_in[6],  (const float*)d_in[12]};
    const float* bih[2] = {(const float*)d_in[7],  (const float*)d_in[13]};
    const float* bhh[2] = {(const float*)d_in[8],  (const float*)d_in[14]};

    const int N = in_sizes[0] / DIM;   // 50000
    const int E = in_sizes[1];         // 800000
    const int ND = N * DIM;

    float* buf0 = (float*)d_ws;
    float* buf1 = buf0 + ND;
    float* mbuf = buf1 + ND;
    float* agg  = mbuf + ND;
    float* bufs[2] = {buf0, buf1};

    const int rowTiles   = N / 16;               // 3125 (N divisible by 16)
    const int zeroBlocks = (ND + 255) / 256;
    const int scatBlocks = (E + 7) / 8;
    const int meanBlocks = (N + 255) / 256;

    const float* hcur = in_feat;
    int nxt = 0;

    for (int layer = 0; layer < 2; ++layer) {
        for (int step = 0; step < 8; ++step) {
            msg_gemm_kernel<<<rowTiles, 256, 0, stream>>>(hcur, Wm[layer], bm[layer], mbuf);
            zero_kernel<<<zeroBlocks, 256, 0, stream>>>(agg, ND);
            scatter_kernel<<<scatBlocks, 256, 0, stream>>>(mbuf, src, dst, agg, E);
            float* hnext = bufs[nxt];
            nxt ^= 1;
            const int do_relu = (layer == 0 && step == 7) ? 1 : 0;
            gru_kernel<<<rowTiles, 256, 0, stream>>>(agg, hcur, Wih[layer], Whh[layer],
                                                     bih[layer], bhh[layer], hnext, do_relu);
            hcur = hnext;
        }
    }

    zero_kernel<<<1, DIM, 0, stream>>>((float*)d_out, DIM);
    mean_kernel<<<meanBlocks, DIM, 0, stream>>>(hcur, (float*)d_out, N, 1.0f / (float)N);
}